// DilatedToothSegmentationNetwork_65953517797784
// MI455X (gfx1250) — compile-verified
//
#include <hip/hip_runtime.h>
#include <hip/hip_bf16.h>
#include <stdint.h>

#define NPTS 10000
#define BNS  0.9999950000374997f   // 1/sqrt(1 + 1e-5)

typedef __attribute__((ext_vector_type(2))) float v2f;
typedef __attribute__((ext_vector_type(8))) float v8f;

// =====================================================================
// WMMA fp32 GEMM:  Y[m, n] = act( BN( X[m,:] . W[n,:] + b[n] ) ) (+ addend)
//   X: M x K (row-major, contiguous), W: Nout x K (row-major)
//   act: 0 = none, 1 = relu, 2 = sigmoid * mulin[m*K+n]   (fi gate, K==Nout)
//   BN applied iff gam != nullptr:  y = gam[n]*y*BNS + bet[n]
//   addend (M x Nout, contiguous) added after activation (residual path)
//   Output written at Y[m*ldY + n]
// One wave computes one 16x16 tile via V_WMMA_F32_16X16X4_F32.
//
// Out-of-range rows/cols are handled by CLAMPING the source row pointer
// (not by zero-fill): garbage in A row m only affects D row m, garbage in
// B col n only affects D col n, and the epilogue never stores rows >= M or
// cols >= Nout. This keeps the inner loop branch-free: two global_load_b64
// plus one v_wmma per K-step, software-pipelined.
// =====================================================================
__global__ __launch_bounds__(256) void dtsn_gemm(
    const float* __restrict__ X, int Kdim,
    const float* __restrict__ W, const float* __restrict__ bias,
    const float* __restrict__ gam, const float* __restrict__ bet,
    const float* __restrict__ addend, const float* __restrict__ mulin,
    float* __restrict__ Y, int ldY, int M, int Nout, int act)
{
  const int lane = threadIdx.x & 31;
  const int wave = threadIdx.x >> 5;
  const int m0 = blockIdx.x * 16;
  const int n0 = (blockIdx.y * 8 + wave) * 16;
  if (n0 >= Nout) return;                       // wave-uniform exit

  const int l16  = lane & 15;
  const int hk   = (lane >> 4) << 1;            // 0 or 2
  const int arow = min(m0 + l16, M - 1);        // clamped: always a valid row
  const int bcol = min(n0 + l16, Nout - 1);     // clamped: always a valid row of W
  const float* __restrict__ Xr = X + (size_t)arow * Kdim + hk;
  const float* __restrict__ Wr = W + (size_t)bcol * Kdim + hk;

  v8f acc = {0.f, 0.f, 0.f, 0.f, 0.f, 0.f, 0.f, 0.f};

  const int kmain = Kdim & ~3;                  // multiple-of-4 body (all K here)
  if (kmain > 0) {
    // software pipeline: fetch (k+4) while multiplying k
    v2f a = *(const v2f*)(Xr);
    v2f b = *(const v2f*)(Wr);
#pragma unroll 4
    for (int k = 4; k < kmain; k += 4) {
      const v2f an = *(const v2f*)(Xr + k);
      const v2f bn = *(const v2f*)(Wr + k);
      acc = __builtin_amdgcn_wmma_f32_16x16x4_f32(false, a, false, b,
                                                  (short)0, acc, false, false);
      a = an; b = bn;
    }
    acc = __builtin_amdgcn_wmma_f32_16x16x4_f32(false, a, false, b,
                                                (short)0, acc, false, false);
  }
  if (kmain < Kdim) {                           // K % 4 tail (unused in this net)
    const int ka = kmain;                       // Xr/Wr already include +hk
    v2f A = {0.f, 0.f}, B = {0.f, 0.f};
    if (ka + hk < Kdim)     { A.x = Xr[ka];     B.x = Wr[ka]; }
    if (ka + hk + 1 < Kdim) { A.y = Xr[ka + 1]; B.y = Wr[ka + 1]; }
    acc = __builtin_amdgcn_wmma_f32_16x16x4_f32(false, A, false, B, (short)0, acc, false, false);
  }

  // D layout: VGPR v -> row m0 + v + 8*(lane>=16), col n0 + (lane&15)
  const int rbase = m0 + ((lane >> 4) << 3);
  const int c = n0 + l16;
#pragma unroll
  for (int v = 0; v < 8; ++v) {
    const int r = rbase + v;
    if (r < M && c < Nout) {
      float y = acc[v];
      if (bias) y += bias[c];
      if (gam)  y = gam[c] * y * BNS + bet[c];
      if (act == 1)      y = fmaxf(y, 0.f);
      else if (act == 2) y = (1.f / (1.f + __expf(-y))) * mulin[(size_t)r * Kdim + c];
      if (addend) y += addend[(size_t)r * Nout + c];
      Y[(size_t)r * ldY + c] = y;
    }
  }
}

// =====================================================================
// Per-point squared norms of feature rows (strided input)
// =====================================================================
__global__ void dtsn_norms(const float* __restrict__ feat, int ldf, int C, int N,
                           float* __restrict__ out)
{
  int i = blockIdx.x * blockDim.x + threadIdx.x;
  if (i >= N) return;
  const float* f = feat + (size_t)i * ldf;
  float s = 0.f;
  for (int c = 0; c < C; ++c) s += f[c] * f[c];
  out[i] = s;
}

// =====================================================================
// Per-row neighbor sort: one workgroup per query row.
// Computes the full distance row (||xi||^2 - 2 xi.xj + ||xj||^2) and bitonic-
// sorts all 10000 (padded to 16384) candidates in 128 KB of LDS (CDNA5:
// up to 320 KB LDS per workgroup). Emits the 32-NN and, optionally, the
// three dilated index sets (stride 6 of top-200, 28 of top-900, 56 of top-1800).
// =====================================================================
#define SORT_M 16384
__global__ __launch_bounds__(1024) void dtsn_knn_sort(
    const float* __restrict__ feat, int ldf, int C,
    const float* __restrict__ norms, int N,
    int* __restrict__ knn_out,
    int* __restrict__ dil1, int* __restrict__ dil2, int* __restrict__ dil3)
{
  __shared__ float skey[SORT_M];
  __shared__ int   sval[SORT_M];
  __shared__ float sx[32];

  const int row = blockIdx.x;
  const int tid = threadIdx.x;

  if (tid < C) sx[tid] = feat[(size_t)row * ldf + tid];
  __syncthreads();

  const float ni = norms[row];
  for (int j = tid; j < SORT_M; j += 1024) {
    if (j < N) {
      const float* fj = feat + (size_t)j * ldf;
      if (j + 1024 < N) __builtin_prefetch(feat + (size_t)(j + 1024) * ldf, 0, 1);
      float dot = 0.f;
      for (int c = 0; c < C; ++c) dot += sx[c] * fj[c];
      skey[j] = ni + norms[j] - 2.f * dot;
      sval[j] = j;
    } else {
      skey[j] = 3.0e38f;
      sval[j] = j;
    }
  }

  for (int size = 2; size <= SORT_M; size <<= 1) {
    for (int stride = size >> 1; stride > 0; stride >>= 1) {
      __syncthreads();
      for (int i = tid; i < SORT_M; i += 1024) {
        const int p = i ^ stride;
        if (p > i) {
          const bool asc = ((i & size) == 0);
          const float ki = skey[i], kp = skey[p];
          if ((ki > kp) == asc) {
            skey[i] = kp; skey[p] = ki;
            const int t = sval[i]; sval[i] = sval[p]; sval[p] = t;
          }
        }
      }
    }
  }
  __syncthreads();

  if (tid < 32) {
    if (knn_out) knn_out[row * 32 + tid] = sval[tid];
    if (dil1)    dil1[row * 32 + tid]    = sval[6  * tid];
    if (dil2)    dil2[row * 32 + tid]    = sval[28 * tid];
    if (dil3)    dil3[row * 32 + tid]    = sval[56 * tid];
  }
}

// =====================================================================
// Edge feature build: E[(p*32+j), :] = [ x[p], x[idx[p][j]] - x[p] ]   (2C wide)
// =====================================================================
__global__ void dtsn_edge_build(const float* __restrict__ x, int ldx, int C,
                                const int* __restrict__ idx,
                                float* __restrict__ E, int N)
{
  const int C2 = 2 * C;
  const unsigned long long total = (unsigned long long)N * 32u * (unsigned)C2;
  for (unsigned long long t = (unsigned long long)blockIdx.x * blockDim.x + threadIdx.x;
       t < total; t += (unsigned long long)gridDim.x * blockDim.x) {
    const int c = (int)(t % (unsigned)C2);
    const unsigned long long rj = t / (unsigned)C2;
    const int j = (int)(rj & 31ull);
    const int p = (int)(rj >> 5);
    float v;
    if (c < C) {
      v = x[(size_t)p * ldx + c];
    } else {
      const int nb = idx[p * 32 + j];
      const int cc = c - C;
      v = x[(size_t)nb * ldx + cc] - x[(size_t)p * ldx + cc];
    }
    E[rj * (unsigned)C2 + (unsigned)c] = v;
  }
}

// =====================================================================
// Max over the 32 neighbors:  Y[p, c] = max_j H[(p*32+j), c]
// =====================================================================
__global__ void dtsn_edge_max(const float* __restrict__ H, int C,
                              float* __restrict__ Y, int ldY, int N)
{
  const unsigned long long total = (unsigned long long)N * (unsigned)C;
  for (unsigned long long t = (unsigned long long)blockIdx.x * blockDim.x + threadIdx.x;
       t < total; t += (unsigned long long)gridDim.x * blockDim.x) {
    const int c = (int)(t % (unsigned)C);
    const int p = (int)(t / (unsigned)C);
    const float* h = H + ((size_t)p * 32) * C + c;
    float m = h[0];
#pragma unroll
    for (int j = 1; j < 32; ++j) m = fmaxf(m, h[(size_t)j * C]);
    Y[(size_t)p * ldY + c] = m;
  }
}

// =====================================================================
// Column-wise max over all rows (STN global pooling):  out[c] = max_r X[r, c]
// =====================================================================
__global__ void dtsn_colmax(const float* __restrict__ X, int M, int C,
                            float* __restrict__ out)
{
  const int c = blockIdx.x * blockDim.x + threadIdx.x;
  if (c >= C) return;
  float m = X[c];
  for (int r = 1; r < M; ++r) m = fmaxf(m, X[(size_t)r * C + c]);
  out[c] = m;
}

// =====================================================================
// Build GEMM weight for x @ t:  Wt[j, k] = t[k, j] = f3v[k*24+j] + (j==k)
// =====================================================================
__global__ void dtsn_make_t(const float* __restrict__ f3v, float* __restrict__ Wt)
{
  const int t = blockIdx.x * blockDim.x + threadIdx.x;
  if (t >= 576) return;
  const int j = t / 24, k = t % 24;
  Wt[t] = f3v[k * 24 + j] + (j == k ? 1.f : 0.f);
}

// =====================================================================
// Host-side wiring
// =====================================================================
struct LinP { const float *W, *b, *g, *bb; };

static inline LinP nextLin(void* const* d, int& i) {
  LinP l;
  l.W  = (const float*)d[i];
  l.b  = (const float*)d[i + 1];
  l.g  = (const float*)d[i + 2];
  l.bb = (const float*)d[i + 3];
  i += 4;
  return l;
}

static inline void gemm(hipStream_t s, const float* X, int K, const LinP& L,
                        bool bn, int act, float* Y, int ldY, int M, int Nout,
                        const float* addend = nullptr, const float* mulin = nullptr)
{
  dim3 g((M + 15) / 16, (Nout + 127) / 128);
  dtsn_gemm<<<g, 256, 0, s>>>(X, K, L.W, L.b, bn ? L.g : nullptr, bn ? L.bb : nullptr,
                              addend, mulin, Y, ldY, M, Nout, act);
}

static inline void edge_conv(hipStream_t s, const float* xin, int ldx, int C,
                             const int* idx, const LinP& l1, const LinP& l2, int Cout,
                             float* E, float* H1, float* H2, float* Y, int ldY)
{
  const int rows = NPTS * 32;
  dtsn_edge_build<<<8192, 256, 0, s>>>(xin, ldx, C, idx, E, NPTS);
  gemm(s, E,  2 * C, l1, true, 1, H1, Cout, rows, Cout);
  gemm(s, H1, Cout,  l2, true, 1, H2, Cout, rows, Cout);
  dtsn_edge_max<<<(NPTS * Cout + 255) / 256, 256, 0, s>>>(H2, Cout, Y, ldY, NPTS);
}

extern "C" void kernel_launch(void* const* d_in, const int* in_sizes, int n_in,
                              void* d_out, int out_size, void* d_ws, size_t ws_size,
                              hipStream_t stream)
{
  const int N = NPTS;
  const float* x   = (const float*)d_in[0];   // (N, 24)
  const float* pos = (const float*)d_in[1];   // (N, 3)

  int i = 2;
  LinP c1 = nextLin(d_in, i), c2 = nextLin(d_in, i), c3 = nextLin(d_in, i);
  LinP f1 = nextLin(d_in, i), f2 = nextLin(d_in, i), f3 = nextLin(d_in, i);
  LinP e1l1 = nextLin(d_in, i), e1l2 = nextLin(d_in, i);
  LinP e2l1 = nextLin(d_in, i), e2l2 = nextLin(d_in, i);
  LinP e3l1 = nextLin(d_in, i), e3l2 = nextLin(d_in, i);
  LinP locl = nextLin(d_in, i);
  LinP d1l1 = nextLin(d_in, i), d1l2 = nextLin(d_in, i);
  LinP d2l1 = nextLin(d_in, i), d2l2 = nextLin(d_in, i);
  LinP d3l1 = nextLin(d_in, i), d3l2 = nextLin(d_in, i);
  LinP glob = nextLin(d_in, i);
  LinP fi   = nextLin(d_in, i);
  LinP r1l1 = nextLin(d_in, i), r1l2 = nextLin(d_in, i), r1rs = nextLin(d_in, i);
  LinP r2l1 = nextLin(d_in, i), r2l2 = nextLin(d_in, i), r2rs = nextLin(d_in, i);
  LinP outl = nextLin(d_in, i);

  // ---- workspace carve-out (bump allocator, 256B aligned) ----
  uint8_t* base = (uint8_t*)d_ws;
  size_t off = 0;
  auto A = [&](size_t bytes) -> void* {
    void* p = base + off;
    off = (off + bytes + 255) & ~(size_t)255;
    return p;
  };
  float* norms  = (float*)A((size_t)N * 4);
  int*   idxK   = (int*)A((size_t)N * 32 * 4);    // 32-NN in pos space (e1)
  int*   idxE   = (int*)A((size_t)N * 32 * 4);    // 32-NN in feature space (e2/e3)
  int*   idxD1  = (int*)A((size_t)N * 32 * 4);
  int*   idxD2  = (int*)A((size_t)N * 32 * 4);
  int*   idxD3  = (int*)A((size_t)N * 32 * 4);
  float* h1     = (float*)A((size_t)N * 64 * 4);
  float* h2     = (float*)A((size_t)N * 128 * 4);
  float* h3     = (float*)A((size_t)N * 1024 * 4);  // reused as glob output later
  float* gvec   = (float*)A(1024 * 4);
  float* f1v    = (float*)A(512 * 4);
  float* f2v    = (float*)A(256 * 4);
  float* f3v    = (float*)A(576 * 4);
  float* tmat   = (float*)A(576 * 4);
  float* xstn   = (float*)A((size_t)N * 24 * 4);
  float* xcat   = (float*)A((size_t)N * 72 * 4);    // [x1 | x2 | x3]
  float* cat240 = (float*)A((size_t)N * 240 * 4);   // [xl | y1 | y2 | y3]
  float* E      = (float*)A((size_t)N * 32 * 120 * 4);  // 153.6 MB, heavily reused
  float* H1buf  = (float*)A((size_t)N * 32 * 60 * 4);
  float* H2buf  = (float*)A((size_t)N * 32 * 60 * 4);

  // Aliases inside dead regions of big buffers:
  float* g1024a = h3;                                       // glob output
  float* g1024b = E;                                        // fi-gated features (41 MB)
  float* r1out  = (float*)((uint8_t*)E + ((size_t)48 << 20));
  float* t256a  = (float*)((uint8_t*)E + ((size_t)72 << 20));
  float* t256b  = (float*)((uint8_t*)E + ((size_t)84 << 20));
  float* r2out  = (float*)((uint8_t*)E + ((size_t)96 << 20));
  float* t512a  = H1buf;
  float* t512b  = H2buf;

  // ================= Phase 0: pos-space neighbor sort ====================
  dtsn_norms<<<(N + 255) / 256, 256, 0, stream>>>(pos, 3, 3, N, norms);
  dtsn_knn_sort<<<N, 1024, 0, stream>>>(pos, 3, 3, norms, N, idxK, idxD1, idxD2, idxD3);

  // ================= Phase 1: STN =======================================
  gemm(stream, x,  24,   c1, true, 1, h1, 64,   N, 64);
  gemm(stream, h1, 64,   c2, true, 1, h2, 128,  N, 128);
  gemm(stream, h2, 128,  c3, true, 1, h3, 1024, N, 1024);
  dtsn_colmax<<<4, 256, 0, stream>>>(h3, N, 1024, gvec);
  gemm(stream, gvec, 1024, f1, true,  1, f1v, 512, 1, 512);
  gemm(stream, f1v,  512,  f2, true,  1, f2v, 256, 1, 256);
  gemm(stream, f2v,  256,  f3, false, 0, f3v, 576, 1, 576);
  dtsn_make_t<<<3, 256, 0, stream>>>(f3v, tmat);
  LinP tlin = {tmat, nullptr, nullptr, nullptr};
  gemm(stream, x, 24, tlin, false, 0, xstn, 24, N, 24);

  // ================= Phase 2: edge convs e1..e3 =========================
  edge_conv(stream, xstn, 24, 24, idxK, e1l1, e1l2, 24, E, H1buf, H2buf, xcat + 0, 72);

  dtsn_norms<<<(N + 255) / 256, 256, 0, stream>>>(xcat + 0, 72, 24, N, norms);
  dtsn_knn_sort<<<N, 1024, 0, stream>>>(xcat + 0, 72, 24, norms, N, idxE,
                                        nullptr, nullptr, nullptr);
  edge_conv(stream, xcat + 0, 72, 24, idxE, e2l1, e2l2, 24, E, H1buf, H2buf, xcat + 24, 72);

  dtsn_norms<<<(N + 255) / 256, 256, 0, stream>>>(xcat + 24, 72, 24, N, norms);
  dtsn_knn_sort<<<N, 1024, 0, stream>>>(xcat + 24, 72, 24, norms, N, idxE,
                                        nullptr, nullptr, nullptr);
  edge_conv(stream, xcat + 24, 72, 24, idxE, e3l1, e3l2, 24, E, H1buf, H2buf, xcat + 48, 72);

  // ================= Phase 3: local + dilated convs =====================
  gemm(stream, xcat, 72, locl, true, 1, cat240 + 0, 240, N, 60);
  edge_conv(stream, cat240 + 0,   240, 60, idxD1, d1l1, d1l2, 60, E, H1buf, H2buf, cat240 + 60,  240);
  edge_conv(stream, cat240 + 60,  240, 60, idxD2, d2l1, d2l2, 60, E, H1buf, H2buf, cat240 + 120, 240);
  edge_conv(stream, cat240 + 120, 240, 60, idxD3, d3l1, d3l2, 60, E, H1buf, H2buf, cat240 + 180, 240);

  // ================= Phase 4: global head ===============================
  gemm(stream, cat240, 240, glob, true, 1, g1024a, 1024, N, 1024);
  // fi gate: g1024b = g1024a * sigmoid(BN(g1024a @ fi.W^T + fi.b))
  gemm(stream, g1024a, 1024, fi, true, 2, g1024b, 1024, N, 1024, nullptr, g1024a);

  // r1: pbr(pbr(x,l1),l2) + bn(pw(x,res))
  gemm(stream, g1024b, 1024, r1l1, true, 1, t512a, 512, N, 512);
  gemm(stream, t512a,  512,  r1l2, true, 1, t512b, 512, N, 512);
  gemm(stream, g1024b, 1024, r1rs, true, 0, r1out, 512, N, 512, t512b);

  // r2
  gemm(stream, r1out, 512, r2l1, true, 1, t256a, 256, N, 256);
  gemm(stream, t256a, 256, r2l2, true, 1, t256b, 256, N, 256);
  gemm(stream, r1out, 512, r2rs, true, 0, r2out, 256, N, 256, t256b);

  // out: plain pw (bias, no BN, no activation)
  gemm(stream, r2out, 256, outl, false, 0, (float*)d_out, 17, N, 17);

  (void)in_sizes; (void)n_in; (void)out_size; (void)ws_size;
}